// GroupedLossWithIndexMap_21131239096802
// MI455X (gfx1250) — compile-verified
//
#include <hip/hip_runtime.h>
#include <math.h>

typedef __attribute__((ext_vector_type(2))) float v2f;
typedef __attribute__((ext_vector_type(8))) float v8f;

// Old-column -> group id, branch-free (group starts: 0,2,4,7,9,12,14,17,19,21).
__device__ __forceinline__ int grp(int k) {
    return (k >= 2) + (k >= 4) + (k >= 7) + (k >= 9) + (k >= 12) +
           (k >= 14) + (k >= 17) + (k >= 19) + (k >= 21);
}

// Extended membership matrix value B[k][col]:
//   col in [0,10): 1 if old column k belongs to group col
//   col == 10    : 1 for every valid k (row-total column)
//   k >= 23      : 0 (K padding)
__device__ __forceinline__ float bval(int k, int col) {
    if (k >= 23) return 0.0f;
    int g = grp(k);
    return (col == g || col == 10) ? 1.0f : 0.0f;
}

// Streaming kernel: each wave consumes `tilesPerWave` consecutive 16-row tiles,
// computes outputs = A(16x24,padded) x Bext(24x16) with 6x V_WMMA_F32_16X16X4_F32,
// normalizes each row by the WMMA-produced total (D column N=10, broadcast via
// ds_bpermute + v_rcp_f32), and accumulates per-lane group partial sums.
// One atomicAdd per (wave, group) at the end.
__global__ void grouped_prob_accum(const float* __restrict__ in,
                                   float* __restrict__ accum,
                                   int tilesPerBatch, int totalTiles,
                                   int tilesPerWave) {
    const int lane = threadIdx.x & 31;
    const int half = lane >> 4;     // K-half for A/B striping, row-half for D
    const int mrow = lane & 15;     // A matrix M row for this lane
    const int col  = lane & 15;     // B/D matrix N column for this lane
    const int wave = (int)((blockIdx.x * (size_t)blockDim.x + threadIdx.x) >> 5);

    int t0 = wave * tilesPerWave;
    if (t0 >= totalTiles) return;            // wave-uniform: EXEC all-1s inside
    int t1 = t0 + tilesPerWave;
    if (t1 > totalTiles) t1 = totalTiles;
    const int b = t0 / tilesPerBatch;        // tilesPerBatch % tilesPerWave == 0

    // Constant B fragments (built once per wave, live in VGPRs).
    v2f bf[6];
#pragma unroll
    for (int ch = 0; ch < 6; ++ch) {
        int k0 = 4 * ch + 2 * half;
        bf[ch].x = bval(k0 + 0, col);
        bf[ch].y = bval(k0 + 1, col);
    }

    const int totSrcLane = (half << 4) + 10;   // lane holding this half's row totals
    float acc = 0.0f;

    for (int t = t0; t < t1; ++t) {
        const float* rowp = in + ((size_t)t * 16 + mrow) * 23 + 2 * half;

        // Unconditional speculative prefetch (global_prefetch_b8, SCOPE_SE):
        // per ISA, translation failures are silently dropped, so running past
        // the end of the buffer on the last tiles is safe. Keeping it branch-
        // free keeps the tile loop uniform (no per-iteration EXEC juggling).
        __builtin_prefetch(rowp + 2 * 16 * 23, 0, 3);

        v8f c = {};
#pragma unroll
        for (int ch = 0; ch < 6; ++ch) {
            v2f a;
            a.x = rowp[4 * ch];
            // K=23 does not exist: zero-pad (also avoids reading 4B past the array)
            a.y = (ch == 5 && half) ? 0.0f : rowp[4 * ch + 1];
            c = __builtin_amdgcn_wmma_f32_16x16x4_f32(
                    /*neg_a=*/false, a, /*neg_b=*/false, bf[ch],
                    /*c_mod=*/(short)0, c, /*reuse_a=*/false, /*reuse_b=*/false);
        }
        // D: lane holds column `col` for rows m = v + 8*half.
        // Row total (B column 10) lives in lane 16*half + 10, same VGPR v.
        // v_rcp_f32 (1-ulp approx TRANS op) instead of the IEEE divide expansion.
#pragma unroll
        for (int v = 0; v < 8; ++v) {
            float tot = __shfl(c[v], totSrcLane, 32);
            acc = fmaf(c[v], __builtin_amdgcn_rcpf(tot), acc);
        }
    }

    // Mask out the total/padding columns once, combine the two row-halves, flush.
    acc *= (col < 10) ? 1.0f : 0.0f;
    acc += __shfl_xor(acc, 16, 32);
    if (lane < 10) atomicAdd(&accum[b * 10 + lane], acc);
}

// Tiny finalize: avg_probs = accum/N; loss = sum_{b,g} t*(log t - log avg) / (B*G).
__global__ void kl_finalize(const float* __restrict__ targets,
                            const float* __restrict__ accum,
                            float* __restrict__ out, float invN) {
    __shared__ float red[128];
    int t = threadIdx.x;
    float v = 0.0f;
    if (t < 80) {
        float tg  = targets[t];
        float avg = accum[t] * invN;
        v = tg * (logf(tg) - logf(avg));
    }
    red[t] = v;
    __syncthreads();
#pragma unroll
    for (int s = 64; s > 0; s >>= 1) {
        if (t < s) red[t] += red[t + s];
        __syncthreads();
    }
    if (t == 0) out[0] = red[0] * (1.0f / 80.0f);
}

extern "C" void kernel_launch(void* const* d_in, const int* in_sizes, int n_in,
                              void* d_out, int out_size, void* d_ws, size_t ws_size,
                              hipStream_t stream) {
    const float* in      = (const float*)d_in[0];   // [8, N, 23] f32
    const float* targets = (const float*)d_in[1];   // [8, 10]    f32
    float* out   = (float*)d_out;                   // scalar
    float* accum = (float*)d_ws;                    // [8, 10] partial prob sums

    const int B = 8, C = 23, G = 10;
    const long long total = in_sizes[0];
    const int N = (int)(total / ((long long)B * C));     // 262144
    const int tilesPerBatch = N / 16;                    // 16384
    const int totalTiles    = B * tilesPerBatch;         // 131072
    const int tilesPerWave  = 16;                        // divides tilesPerBatch
    const int waves   = (totalTiles + tilesPerWave - 1) / tilesPerWave;
    const int threads = 256;                             // 8 waves per block
    const int blocks  = (waves * 32 + threads - 1) / threads;

    hipMemsetAsync(accum, 0, B * G * sizeof(float), stream);
    grouped_prob_accum<<<blocks, threads, 0, stream>>>(
        in, accum, tilesPerBatch, totalTiles, tilesPerWave);
    kl_finalize<<<1, 128, 0, stream>>>(targets, accum, out, 1.0f / (float)N);
}